// QCO_1d_19061064860185
// MI455X (gfx1250) — compile-verified
//
#include <hip/hip_runtime.h>
#include <math.h>

#define LEVEL_NUM 64
#define IN_CH 16
#define WIDTH 128
#define N_IMG 8
#define HDIM 256
#define HW 65536
#define EPSV 1e-5f

typedef float v2f __attribute__((ext_vector_type(2)));
typedef float v8f __attribute__((ext_vector_type(8)));

// ---- workspace layout (float element offsets) ----
#define WS_H1        0                    // 8*16*65536 = 8388608
#define WS_BN1SUM    8388608              // 16
#define WS_BN1SQ     (WS_BN1SUM+16)       // 16
#define WS_BN1SCALE  (WS_BN1SUM+32)       // 16
#define WS_BN1SHIFT  (WS_BN1SUM+48)       // 16
#define WS_ACTMEAN   (WS_BN1SUM+64)       // 8*16
#define WS_XAVE      (WS_ACTMEAN+128)     // 8*128
#define WS_XAH       (WS_XAVE+1024)       // 8*128
#define WS_COS       (WS_XAH+1024)        // 8*65536
#define WS_CMIN      (WS_COS+524288)      // 8 (sortable uint)
#define WS_CMAX      (WS_CMIN+8)          // 8 (sortable uint)
#define WS_STASUM    (WS_CMAX+8)          // 8*64
#define WS_STAPRE    (WS_STASUM+512)      // 8*128*64
#define WS_OBNSUM    (WS_STAPRE+65536)    // 128
#define WS_OBNSQ     (WS_OBNSUM+128)      // 128

__device__ __forceinline__ float gelu_exact(float v){
    return 0.5f * v * (1.0f + erff(v * 0.70710678118654752f));
}
// order-preserving float<->uint mapping for atomic min/max
__device__ __forceinline__ unsigned f2sort(float f){
    unsigned b = __float_as_uint(f);
    return (b & 0x80000000u) ? ~b : (b | 0x80000000u);
}
__device__ __forceinline__ float sort2f(unsigned u){
    unsigned b = (u & 0x80000000u) ? (u & 0x7FFFFFFFu) : ~u;
    return __uint_as_float(b);
}

// ---------------- K0: init small stats regions ----------------
__global__ void k0_init(float* ws){
    int t = threadIdx.x;
    if (t < 64)  ws[WS_BN1SUM + t] = 0.f;          // sums/sq/scale/shift area
    if (t < 128) ws[WS_ACTMEAN + t] = 0.f;
    if (t < 512) ws[WS_STASUM + t] = 0.f;
    if (t < 256) ws[WS_OBNSUM + t] = 0.f;
    unsigned* u = (unsigned*)ws;
    if (t < 8)            u[WS_CMIN + t]     = 0xFFFFFFFFu;
    if (t >= 8 && t < 16) u[WS_CMAX + (t-8)] = 0u;
}

// ---------------- K1: 3x3 conv 16->16, write h1, BN stats ----------------
__global__ __launch_bounds__(256)
void k1_conv1(const float* __restrict__ x, const float* __restrict__ w,
              const float* __restrict__ b, float* __restrict__ ws){
    __shared__ float tile[IN_CH*18*18];
    __shared__ float wl[IN_CH*IN_CH*9];
    __shared__ float bsum[IN_CH], bsq[IN_CH];
    const int tid = threadIdx.y*16 + threadIdx.x;
    const int n = blockIdx.z;
    if (tid < IN_CH){ bsum[tid]=0.f; bsq[tid]=0.f; }
    for (int i = tid; i < IN_CH*IN_CH*9; i += 256) wl[i] = w[i];
    const int x0 = blockIdx.x*16 - 1, y0 = blockIdx.y*16 - 1;
    for (int c = 0; c < IN_CH; c++){
        const float* xp = x + (((size_t)n*IN_CH + c) << 16);
        for (int i = tid; i < 324; i += 256){
            int ly = i / 18, lx = i - ly*18;
            int gy = y0 + ly, gx = x0 + lx;
            float v = 0.f;
            if (gy >= 0 && gy < HDIM && gx >= 0 && gx < HDIM) v = xp[gy*HDIM + gx];
            tile[c*324 + i] = v;
        }
    }
    __syncthreads();
    float acc[IN_CH];
    #pragma unroll
    for (int o = 0; o < IN_CH; o++) acc[o] = b[o];
    for (int c = 0; c < IN_CH; c++){
        #pragma unroll
        for (int ky = 0; ky < 3; ky++){
            #pragma unroll
            for (int kx = 0; kx < 3; kx++){
                float v = tile[c*324 + (threadIdx.y+ky)*18 + threadIdx.x+kx];
                #pragma unroll
                for (int o = 0; o < IN_CH; o++)
                    acc[o] = fmaf(wl[o*144 + c*9 + ky*3 + kx], v, acc[o]);
            }
        }
    }
    const int gy = blockIdx.y*16 + threadIdx.y;
    const int gx = blockIdx.x*16 + threadIdx.x;
    #pragma unroll
    for (int o = 0; o < IN_CH; o++)
        ws[WS_H1 + (((size_t)n*IN_CH + o) << 16) + gy*HDIM + gx] = acc[o];
    #pragma unroll
    for (int o = 0; o < IN_CH; o++){
        float s = acc[o], q = acc[o]*acc[o];
        #pragma unroll
        for (int off = 16; off >= 1; off >>= 1){
            s += __shfl_xor(s, off, 32);
            q += __shfl_xor(q, off, 32);
        }
        if ((tid & 31) == 0){ atomicAdd(&bsum[o], s); atomicAdd(&bsq[o], q); }
    }
    __syncthreads();
    if (tid < IN_CH){
        atomicAdd(&ws[WS_BN1SUM + tid], bsum[tid]);
        atomicAdd(&ws[WS_BN1SQ  + tid], bsq[tid]);
    }
}

// ---------------- K2: finalize BN1 scale/shift ----------------
__global__ void k2_bn1(const float* __restrict__ g, const float* __restrict__ b, float* ws){
    int t = threadIdx.x;
    if (t >= IN_CH) return;
    const float inv = 1.0f / (float)((size_t)N_IMG*HW);
    float mu  = ws[WS_BN1SUM + t] * inv;
    float var = ws[WS_BN1SQ  + t] * inv - mu*mu;
    float sc  = g[t] * rsqrtf(var + EPSV);
    ws[WS_BN1SCALE + t] = sc;
    ws[WS_BN1SHIFT + t] = b[t] - mu*sc;
}

// ---------------- K3: per-(n,c) mean of GELU(BN(h1)) ----------------
__global__ __launch_bounds__(256)
void k3_actmean(float* ws){
    __shared__ float red[8];
    const int n = blockIdx.z, c = blockIdx.y, t = threadIdx.x;
    const float sc = ws[WS_BN1SCALE + c], sh = ws[WS_BN1SHIFT + c];
    const float* h1 = ws + WS_H1 + (((size_t)n*IN_CH + c) << 16);
    const int base = blockIdx.x * 1024;
    float s = 0.f;
    #pragma unroll
    for (int j = 0; j < 4; j++)
        s += gelu_exact(fmaf(h1[base + j*256 + t], sc, sh));
    #pragma unroll
    for (int off = 16; off >= 1; off >>= 1) s += __shfl_xor(s, off, 32);
    if ((t & 31) == 0) red[t >> 5] = s;
    __syncthreads();
    if (t == 0){
        float tot = 0.f;
        #pragma unroll
        for (int i = 0; i < 8; i++) tot += red[i];
        atomicAdd(&ws[WS_ACTMEAN + n*IN_CH + c], tot);
    }
}

// ---------------- K4: x_ave = conv2(actmean)+b2, and its l2norm ----------------
__global__ __launch_bounds__(1024)
void k4_xave(const float* __restrict__ w2, const float* __restrict__ b2, float* ws){
    __shared__ float nrm[N_IMG];
    const int t = threadIdx.x;
    const int o = t & (WIDTH-1), n = t >> 7;
    if (t < N_IMG) nrm[t] = 0.f;
    __syncthreads();
    float v = b2[o];
    const float inv = 1.0f / (float)HW;
    #pragma unroll
    for (int c = 0; c < IN_CH; c++)
        v = fmaf(w2[o*IN_CH + c], ws[WS_ACTMEAN + n*IN_CH + c] * inv, v);
    ws[WS_XAVE + n*WIDTH + o] = v;
    atomicAdd(&nrm[n], v*v);
    __syncthreads();
    ws[WS_XAH + n*WIDTH + o] = v / fmaxf(sqrtf(nrm[n]), 1e-12f);
}

// ---------------- K5: WMMA conv2 recompute + cos_sim + min/max ----------------
// grid (128, 8) x 256 threads; each wave does 4 tiles of 16 pixels.
__global__ __launch_bounds__(256)
void k5_cos_wmma(const float* __restrict__ w2, const float* __restrict__ b2, float* ws){
    __shared__ float xah_s[WIDTH];
    __shared__ float scs[IN_CH], shs[IN_CH];
    __shared__ unsigned smin, smax;
    const int tid = threadIdx.x;
    const int n = blockIdx.y;
    if (tid < WIDTH) xah_s[tid] = ws[WS_XAH + n*WIDTH + tid];
    if (tid < IN_CH){ scs[tid] = ws[WS_BN1SCALE + tid]; shs[tid] = ws[WS_BN1SHIFT + tid]; }
    if (tid == 0){ smin = 0xFFFFFFFFu; smax = 0u; }
    __syncthreads();
    const int wave = tid >> 5, lane = tid & 31;
    const int half = lane >> 4, sub = lane & 15;
    // Preload B (weights, 4x16 K x N tiles) and bias, per ISA f32 operand layout.
    v2f  bw[8][4];
    float barr[8];
    #pragma unroll
    for (int g = 0; g < 8; g++){
        #pragma unroll
        for (int k = 0; k < 4; k++){
            const int row = g*16 + sub;       // N = output channel
            const int kk  = 4*k + half*2;     // K index pair base
            bw[g][k].x = w2[row*IN_CH + kk];
            bw[g][k].y = w2[row*IN_CH + kk + 1];
        }
        barr[g] = b2[g*16 + sub];
    }
    const float* h1 = ws + WS_H1 + (((size_t)n*IN_CH) << 16);
    float* cosp = ws + WS_COS + (size_t)n*HW;
    unsigned lmin = 0xFFFFFFFFu, lmax = 0u;
    for (int tIdx = 0; tIdx < 4; tIdx++){
        const int base = ((blockIdx.x*8 + wave)*4 + tIdx) * 16;
        // A: 16 px x 4 in-ch activation tiles (recomputed BN+GELU)
        v2f A[4];
        #pragma unroll
        for (int k = 0; k < 4; k++){
            const int cc = 4*k + half*2;
            const int px = base + sub;
            A[k].x = gelu_exact(fmaf(h1[(cc    << 16) + px], scs[cc],   shs[cc]));
            A[k].y = gelu_exact(fmaf(h1[((cc+1)<< 16) + px], scs[cc+1], shs[cc+1]));
        }
        float dotp[8], sq[8];
        #pragma unroll
        for (int v = 0; v < 8; v++){ dotp[v] = 0.f; sq[v] = 0.f; }
        #pragma unroll
        for (int g = 0; g < 8; g++){
            v8f acc;
            #pragma unroll
            for (int v = 0; v < 8; v++) acc[v] = barr[g];
            #pragma unroll
            for (int k = 0; k < 4; k++)
                acc = __builtin_amdgcn_wmma_f32_16x16x4_f32(
                        false, A[k], false, bw[g][k], (short)0, acc, false, false);
            const float xv = xah_s[g*16 + sub];
            #pragma unroll
            for (int v = 0; v < 8; v++){
                dotp[v] = fmaf(xv, acc[v], dotp[v]);
                sq[v]   = fmaf(acc[v], acc[v], sq[v]);
            }
        }
        // reduce over the 16 channel lanes within each half-wave
        #pragma unroll
        for (int v = 0; v < 8; v++){
            #pragma unroll
            for (int off = 1; off < 16; off <<= 1){
                dotp[v] += __shfl_xor(dotp[v], off, 16);
                sq[v]   += __shfl_xor(sq[v],   off, 16);
            }
        }
        if (sub < 8){
            float cs = dotp[sub] / fmaxf(sqrtf(sq[sub]), 1e-12f);
            cosp[base + half*8 + sub] = cs;
            unsigned u = f2sort(cs);
            lmin = min(lmin, u); lmax = max(lmax, u);
        }
    }
    atomicMin(&smin, lmin);
    atomicMax(&smax, lmax);
    __syncthreads();
    if (tid == 0){
        unsigned* uws = (unsigned*)ws;
        atomicMin(&uws[WS_CMIN + n], smin);
        atomicMax(&uws[WS_CMAX + n], smax);
    }
}

// ---------------- K6: quant writes + sparse sta accumulation ----------------
__global__ __launch_bounds__(256)
void k6_quant(float* __restrict__ ws, float* __restrict__ outq){
    __shared__ float s_c[256];
    __shared__ float s_sta[LEVEL_NUM];
    const int t = threadIdx.x, n = blockIdx.y;
    if (t < LEVEL_NUM) s_sta[t] = 0.f;
    const unsigned* uws = (const unsigned*)ws;
    const float cmin = sort2f(uws[WS_CMIN + n]);
    const float cmax = sort2f(uws[WS_CMAX + n]);
    const float qint = (cmax - cmin) * (1.0f / LEVEL_NUM);
    const int pxbase = blockIdx.x * 256;
    const float c = ws[WS_COS + (size_t)n*HW + pxbase + t];
    s_c[t] = c;
    __syncthreads();
    if (qint > 0.f){
        // at most 2 adjacent levels pass the threshold |ql - c| < qint
        float z = (c - cmin) / qint;
        int l0 = (int)floorf(z - 0.5f);
        #pragma unroll
        for (int j = 0; j < 2; j++){
            int l = l0 + j;
            if (l >= 0 && l < LEVEL_NUM){
                float ql = fmaf((float)l + 0.5f, qint, cmin);
                float q  = 1.0f - fabsf(ql - c);
                if (q > 1.0f - qint) atomicAdd(&s_sta[l], q);
            }
        }
    }
    // dense quant writes, float4-vectorized, fully coalesced
    float4* qout = (float4*)(outq + ((size_t)n*HW + pxbase) * LEVEL_NUM);
    for (int it = 0; it < 16; it++){
        int idx = it*256 + t;
        int pi = idx >> 4, ch = idx & 15;
        float cc = s_c[pi];
        float rr[4];
        #pragma unroll
        for (int j = 0; j < 4; j++){
            int lv = ch*4 + j;
            float ql = fmaf((float)lv + 0.5f, qint, cmin);
            float q  = 1.0f - fabsf(ql - cc);
            rr[j] = (q > 1.0f - qint) ? q : 0.f;
        }
        qout[(size_t)pi*16 + ch] = make_float4(rr[0], rr[1], rr[2], rr[3]);
    }
    __syncthreads();
    if (t < LEVEL_NUM) atomicAdd(&ws[WS_STASUM + n*LEVEL_NUM + t], s_sta[t]);
}

// ---------------- K7a: head (f1, f2, concat ave, out matmul) + BN1d stats ----------------
__global__ __launch_bounds__(128)
void k7a_head(const float* __restrict__ f1w, const float* __restrict__ f1b,
              const float* __restrict__ f2w, const float* __restrict__ f2b,
              const float* __restrict__ ow,  const float* __restrict__ ob,
              float* ws){
    __shared__ float A64[64], B128[WIDTH], X[WIDTH];
    __shared__ float tot;
    const int t = threadIdx.x;
    const int n = blockIdx.x >> 6;
    const int l = blockIdx.x & 63;
    const unsigned* uws = (const unsigned*)ws;
    const float cmin = sort2f(uws[WS_CMIN + n]);
    const float cmax = sort2f(uws[WS_CMAX + n]);
    const float qint = (cmax - cmin) * (1.0f / LEVEL_NUM);
    const float ql = fmaf((float)l + 0.5f, qint, cmin);
    if (t == 0){
        float s = 0.f;
        for (int i = 0; i < LEVEL_NUM; i++) s += ws[WS_STASUM + n*LEVEL_NUM + i];
        tot = s;
    }
    X[t] = ws[WS_XAVE + n*WIDTH + t];
    __syncthreads();
    const float stan = ws[WS_STASUM + n*LEVEL_NUM + l] / tot;
    if (t < 64)
        A64[t] = fmaxf(fmaf(f1w[t*2], ql, fmaf(f1w[t*2+1], stan, f1b[t])), 0.f);
    __syncthreads();
    float f2 = f2b[t];
    #pragma unroll 8
    for (int c2 = 0; c2 < 64; c2++) f2 = fmaf(f2w[t*64 + c2], A64[c2], f2);
    B128[t] = fmaxf(f2, 0.f);
    __syncthreads();
    float pre = ob[t];
    #pragma unroll 8
    for (int c2 = 0; c2 < WIDTH; c2++) pre = fmaf(ow[t*256 + c2],        B128[c2], pre);
    #pragma unroll 8
    for (int c2 = 0; c2 < WIDTH; c2++) pre = fmaf(ow[t*256 + WIDTH + c2], X[c2],   pre);
    ws[WS_STAPRE + ((size_t)n*WIDTH + t)*LEVEL_NUM + l] = pre;
    atomicAdd(&ws[WS_OBNSUM + t], pre);
    atomicAdd(&ws[WS_OBNSQ  + t], pre*pre);
}

// ---------------- K7b: BN1d + ReLU -> sta output ----------------
__global__ __launch_bounds__(256)
void k7b_bnout(const float* __restrict__ g, const float* __restrict__ b,
               const float* __restrict__ ws, float* __restrict__ outsta){
    const int idx = blockIdx.x*256 + threadIdx.x;        // 65536 elems
    const int o = (idx >> 6) & (WIDTH-1);
    const float inv = 1.0f / (float)(N_IMG*LEVEL_NUM);
    float mu  = ws[WS_OBNSUM + o] * inv;
    float var = ws[WS_OBNSQ  + o] * inv - mu*mu;
    float v = (ws[WS_STAPRE + idx] - mu) * rsqrtf(var + EPSV) * g[o] + b[o];
    outsta[idx] = fmaxf(v, 0.f);
}

extern "C" void kernel_launch(void* const* d_in, const int* in_sizes, int n_in,
                              void* d_out, int out_size, void* d_ws, size_t ws_size,
                              hipStream_t stream){
    const float* x       = (const float*)d_in[0];
    const float* conv1_w = (const float*)d_in[1];
    const float* conv1_b = (const float*)d_in[2];
    const float* bn1_g   = (const float*)d_in[3];
    const float* bn1_b   = (const float*)d_in[4];
    const float* conv2_w = (const float*)d_in[5];
    const float* conv2_b = (const float*)d_in[6];
    const float* f1_w    = (const float*)d_in[7];
    const float* f1_b    = (const float*)d_in[8];
    const float* f2_w    = (const float*)d_in[9];
    const float* f2_b    = (const float*)d_in[10];
    const float* out_w   = (const float*)d_in[11];
    const float* out_b   = (const float*)d_in[12];
    const float* obn_g   = (const float*)d_in[13];
    const float* obn_b   = (const float*)d_in[14];

    float* ws  = (float*)d_ws;
    float* out = (float*)d_out;
    float* out_sta   = out;                       // [8,128,64]
    float* out_quant = out + (size_t)N_IMG*WIDTH*LEVEL_NUM; // [8,65536,64]

    k0_init<<<1, 1024, 0, stream>>>(ws);
    k1_conv1<<<dim3(16,16,N_IMG), dim3(16,16), 0, stream>>>(x, conv1_w, conv1_b, ws);
    k2_bn1<<<1, 32, 0, stream>>>(bn1_g, bn1_b, ws);
    k3_actmean<<<dim3(64,IN_CH,N_IMG), 256, 0, stream>>>(ws);
    k4_xave<<<1, 1024, 0, stream>>>(conv2_w, conv2_b, ws);
    k5_cos_wmma<<<dim3(128,N_IMG), 256, 0, stream>>>(conv2_w, conv2_b, ws);
    k6_quant<<<dim3(256,N_IMG), 256, 0, stream>>>(ws, out_quant);
    k7a_head<<<N_IMG*LEVEL_NUM, WIDTH, 0, stream>>>(f1_w, f1_b, f2_w, f2_b, out_w, out_b, ws);
    k7b_bnout<<<256, 256, 0, stream>>>(obn_g, obn_b, ws, out_sta);
}